// IsoVelo_DNN_module_84516366451206
// MI455X (gfx1250) — compile-verified
//
#include <hip/hip_runtime.h>

// ---------------------------------------------------------------------------
// IsoVelo cosine-similarity cost, MI455X (gfx1250, wave32).
//
// One wave per cell. dots[k] = v . vn[k] on the matrix pipe via
// V_WMMA_F32_16X16X4_F32 (exact f32): A = v broadcast over M rows,
// B = VN^T (column n = neighbor n), all D rows equal -> acc[0] = dot[col].
//
// Each lane materializes ONLY its own WMMA fragment elements straight from
// memory (lane-dependent addresses, no dynamically-indexed private arrays),
// so no v_cndmask select chains. Norms are computed as lane-pair partials
// combined with shfl_xor(,16): the two lane halves own disjoint dim sets.
// ---------------------------------------------------------------------------

#define N_CELLS 100000
#define N_ISO   16
#define KP1     31      // indices row length (col 0 unused)
#define NK      30      // neighbors per cell
#define NEG_INF (-3.402823466e38f)

typedef __attribute__((ext_vector_type(2))) float v2f;
typedef __attribute__((ext_vector_type(8))) float v8f;

__global__ __launch_bounds__(256)
void isovelo_cost_kernel(const float* __restrict__ unsplice,
                         const float* __restrict__ splices,
                         const float* __restrict__ unsplice_predict,
                         const float* __restrict__ splice_predicts,
                         const int*   __restrict__ indices,
                         float*       __restrict__ cost)
{
    const int lane = threadIdx.x & 31;
    const int wave = threadIdx.x >> 5;
    const int cell = blockIdx.x * 8 + wave;        // 12500 blocks * 8 waves = 100000
    if (cell >= N_CELLS) return;                   // wave-uniform (never taken here)

    const int col  = lane & 15;                    // WMMA column N
    const int hi   = lane >> 4;                    // lane half: selects K pair
    const int base = hi * 2;

    // Iso-array offsets (j = dim-1) for this lane's fragment slots.
    // dim(kk,s) = 4kk + base + s; specials: (0,0) dim 0 uses unsplice (hi=0),
    // (4,0) is dim 16 (hi=0) or padding-zero (hi=1); (4,1) is always zero.
    const int j00 = hi ? 1 : 0;                    // (0,0) when hi=1 -> dim2 -> j=1
    const int jA  = base;                          // (0,1): dim base+1
    const int j10 = 3 + base,  j11 = 4 + base;     // kk=1
    const int j20 = 7 + base,  j21 = 8 + base;     // kk=2
    const int j30 = 11 + base, j31 = 12 + base;    // kk=3
    const int j40 = hi ? 0 : 15;                   // (4,0) when hi=0 -> dim16 -> j=15

    const float u0 = unsplice[cell];
    const float up = unsplice_predict[cell];

    const float* __restrict__ spc = splices         + (long)cell * N_ISO;
    const float* __restrict__ ppc = splice_predicts + (long)cell * N_ISO;

    // own splice values at this lane's offsets (shared by v and every vn)
    const float so00 = spc[j00], soA  = spc[jA];
    const float so10 = spc[j10], so11 = spc[j11];
    const float so20 = spc[j20], so21 = spc[j21];
    const float so30 = spc[j30], so31 = spc[j31];
    const float so40 = spc[j40];

    // ---- v fragment (A matrix, broadcast over M rows) ---------------------
    const float v0  = up - u0;
    const float a00 = hi ? (ppc[j00] - so00) : v0;
    const float a01 = ppc[jA]  - soA;
    const float a10 = ppc[j10] - so10, a11 = ppc[j11] - so11;
    const float a20 = ppc[j20] - so20, a21 = ppc[j21] - so21;
    const float a30 = ppc[j30] - so30, a31 = ppc[j31] - so31;
    const float a40 = hi ? 0.0f : (ppc[j40] - so40);

    float vpart = a00 * a00 + a01 * a01 + a10 * a10 + a11 * a11 + a20 * a20
                + a21 * a21 + a30 * a30 + a31 * a31 + a40 * a40;
    const float vnorm2 = vpart + __shfl_xor(vpart, 16, 32);
    const float vnorm  = sqrtf(vnorm2);

    const v2f A0 = {a00, a01};
    const v2f A1 = {a10, a11};
    const v2f A2 = {a20, a21};
    const v2f A3 = {a30, a31};
    const v2f A4 = {a40, 0.0f};

    float best = NEG_INF;

#pragma unroll
    for (int chain = 0; chain < 2; ++chain) {
        const int  k     = chain * 16 + col;       // neighbor ordinal 0..31
        const bool valid = (k < NK);
        const int  nbr   = valid ? indices[cell * KP1 + 1 + k] : 0;

        const float* __restrict__ spn = splices + (long)nbr * N_ISO;

        // ---- vn fragment (B matrix, column = col) -------------------------
        const float n0  = unsplice[nbr] - u0;
        const float b00 = hi ? (spn[j00] - so00) : n0;
        const float b01 = spn[jA]  - soA;
        const float b10 = spn[j10] - so10, b11 = spn[j11] - so11;
        const float b20 = spn[j20] - so20, b21 = spn[j21] - so21;
        const float b30 = spn[j30] - so30, b31 = spn[j31] - so31;
        const float b40 = hi ? 0.0f : (spn[j40] - so40);

        float npart = b00 * b00 + b01 * b01 + b10 * b10 + b11 * b11 + b20 * b20
                    + b21 * b21 + b30 * b30 + b31 * b31 + b40 * b40;
        const float nnorm2 = npart + __shfl_xor(npart, 16, 32);

        const v2f B0 = {b00, b01};
        const v2f B1 = {b10, b11};
        const v2f B2 = {b20, b21};
        const v2f B3 = {b30, b31};
        const v2f B4 = {b40, 0.0f};

        // ---- WMMA chain: 5 K-steps of 16x16x4 f32 -------------------------
        v8f acc = {};
        acc = __builtin_amdgcn_wmma_f32_16x16x4_f32(false, A0, false, B0, (short)0, acc, false, false);
        acc = __builtin_amdgcn_wmma_f32_16x16x4_f32(false, A1, false, B1, (short)0, acc, false, false);
        acc = __builtin_amdgcn_wmma_f32_16x16x4_f32(false, A2, false, B2, (short)0, acc, false, false);
        acc = __builtin_amdgcn_wmma_f32_16x16x4_f32(false, A3, false, B3, (short)0, acc, false, false);
        acc = __builtin_amdgcn_wmma_f32_16x16x4_f32(false, A4, false, B4, (short)0, acc, false, false);
        const float dot = acc[0];                  // all rows equal (A broadcast)

        float denom = vnorm * sqrtf(nnorm2);
        denom = (denom == 0.0f) ? 1.0f : denom;    // reference: denom==0 -> 1
        float cs = dot / denom;
        if (!valid) cs = NEG_INF;                  // mask padding columns
        best = fmaxf(best, cs);
    }

    // wave32 max reduction (lane pairs hold duplicate values: harmless)
#pragma unroll
    for (int off = 16; off >= 1; off >>= 1)
        best = fmaxf(best, __shfl_xor(best, off, 32));

    if (lane == 0) cost[cell] = 1.0f - best;
}

__global__ __launch_bounds__(256)
void isovelo_reduce_kernel(const float* __restrict__ cost,
                           float*       __restrict__ out)
{
    __shared__ float sm[256];
    float s = 0.0f;
    for (int i = (int)threadIdx.x; i < N_CELLS; i += 256)  // fixed order: deterministic
        s += cost[i];
    sm[threadIdx.x] = s;
    __syncthreads();
    for (int off = 128; off >= 1; off >>= 1) {
        if ((int)threadIdx.x < off) sm[threadIdx.x] += sm[threadIdx.x + off];
        __syncthreads();
    }
    if (threadIdx.x == 0) out[0] = sm[0] / (float)N_CELLS;
}

extern "C" void kernel_launch(void* const* d_in, const int* in_sizes, int n_in,
                              void* d_out, int out_size, void* d_ws, size_t ws_size,
                              hipStream_t stream)
{
    const float* unsplice         = (const float*)d_in[0];
    const float* splices          = (const float*)d_in[1];
    const float* unsplice_predict = (const float*)d_in[2];
    const float* splice_predicts  = (const float*)d_in[3];
    const int*   indices          = (const int*)d_in[4];

    float* cost = (float*)d_ws;                    // 100000 floats of scratch

    isovelo_cost_kernel<<<N_CELLS / 8, 256, 0, stream>>>(
        unsplice, splices, unsplice_predict, splice_predicts, indices, cost);
    isovelo_reduce_kernel<<<1, 256, 0, stream>>>(cost, (float*)d_out);
}